// PointsGeneration_83708912599167
// MI455X (gfx1250) — compile-verified
//
#include <hip/hip_runtime.h>
#include <stdint.h>

// Elementwise streaming kernel for PointsGeneration head.
// HBM-bound: 512MB total traffic -> ~22us floor @ 23.3 TB/s.
// Inputs staged via gfx1250 async global->LDS DMA (ASYNCcnt-tracked),
// double-buffered; outputs via non-temporal b128 stores.

#define HW_TOTAL (2048u * 2048u)   // pixels per plane
#define TILE_PIX 512u              // pixels per tile per block
#define NPLANES 16u                // 4 score planes + 12 reg planes
#define THREADS 128u               // 4 waves (wave32)
#define TPT 4u                     // pixels per thread (one float4)
#define TILES_PER_BLOCK 4u
#define NBLOCKS 2048u              // 2048 * 4 * 512 = 4M pixels exactly

// logit(0.45) = ln(0.45/0.55); sigmoid(x) > 0.45  <=>  x > logit(0.45)
#define LOGIT_TEXT (-0.2006706954621511f)

typedef float v4f __attribute__((ext_vector_type(4)));

__device__ __forceinline__ float fast_sigmoid(float x) {
  // v_exp_f32 + v_rcp_f32 path (no correctness harness; bandwidth-bound anyway)
  return __builtin_amdgcn_rcpf(1.0f + __expf(-x));
}

// Issue 16 async b128 global->LDS loads (one per plane) for one tile.
// Each lane moves its own 16 bytes per plane; per-wave ASYNCcnt += 16.
__device__ __forceinline__ void issue_tile(const uint64_t* bases, uint32_t tile,
                                           uint32_t buf, uint32_t tid,
                                           uint32_t lds0) {
  const uint32_t voff = (tile * TILE_PIX + tid * TPT) * 4u;  // byte off in plane
#pragma unroll
  for (uint32_t p = 0; p < NPLANES; ++p) {
    const uint32_t lds = lds0 + ((buf * NPLANES + p) * TILE_PIX + tid * TPT) * 4u;
    asm volatile("global_load_async_to_lds_b128 %0, %1, %2"
                 :
                 : "v"(lds), "v"(voff), "s"(bases[p])
                 : "memory");
  }
}

__device__ __forceinline__ void compute_tile(const float* sm, uint32_t buf,
                                             uint32_t tile, uint32_t tid,
                                             float* __restrict__ out) {
  const float* b = sm + (size_t)buf * NPLANES * TILE_PIX;
  const uint32_t o = tid * TPT;

  v4f x[4];   // raw score logits: text, head, tail, bond
#pragma unroll
  for (int p = 0; p < 4; ++p) x[p] = *(const v4f*)(b + (size_t)p * TILE_PIX + o);
  v4f r[12];  // reg_head[0..3], reg_tail[0..3], reg_bond[0..3]
#pragma unroll
  for (int p = 0; p < 12; ++p)
    r[p] = *(const v4f*)(b + (size_t)(4 + p) * TILE_PIX + o);

  v4f sg[4], mh, ml, mb;
#pragma unroll
  for (int j = 0; j < 4; ++j) {
    const float t = x[0][j], h = x[1][j], l = x[2][j], bo = x[3][j];
    sg[0][j] = fast_sigmoid(t);
    sg[1][j] = fast_sigmoid(h);
    sg[2][j] = fast_sigmoid(l);
    sg[3][j] = fast_sigmoid(bo);
    // thresholds on raw logits: sigmoid(v) > thr  <=>  v > logit(thr)
    const float mt = (t > LOGIT_TEXT) ? 1.0f : 0.0f;
    mh[j] = (h > 0.0f) ? mt : 0.0f;   // head gated by text mask
    ml[j] = (l > 0.0f) ? mt : 0.0f;   // tail gated by text mask
    mb[j] = (bo > 0.0f) ? 1.0f : 0.0f;
  }

  const size_t pix = (size_t)tile * TILE_PIX + o;
#pragma unroll
  for (int p = 0; p < 4; ++p)
    __builtin_nontemporal_store(sg[p], (v4f*)(out + (size_t)p * HW_TOTAL + pix));
#pragma unroll
  for (int c = 0; c < 4; ++c) {
    __builtin_nontemporal_store(r[c] * mh,
                                (v4f*)(out + (size_t)(4 + c) * HW_TOTAL + pix));
    __builtin_nontemporal_store(r[4 + c] * ml,
                                (v4f*)(out + (size_t)(8 + c) * HW_TOTAL + pix));
    __builtin_nontemporal_store(r[8 + c] * mb,
                                (v4f*)(out + (size_t)(12 + c) * HW_TOTAL + pix));
  }
}

__global__ __launch_bounds__(THREADS) void points_gen_stream(
    const float* __restrict__ st, const float* __restrict__ sh,
    const float* __restrict__ sl, const float* __restrict__ sb,
    const float* __restrict__ rh, const float* __restrict__ rt,
    const float* __restrict__ rb, float* __restrict__ out) {
  // 2 x (16 planes x 512 px x 4B) = 64KB double-buffered stage
  __shared__ __align__(16) float smem[2 * NPLANES * TILE_PIX];
  const uint32_t tid = threadIdx.x;
  // Generic LDS addresses carry the LDS byte offset in the low 32 bits
  // (ISA 10.2 aperture mapping) -> valid VDST for async-to-LDS ops.
  const uint32_t lds0 = (uint32_t)(size_t)(&smem[0]);

  uint64_t bases[NPLANES];
  bases[0] = (uint64_t)(uintptr_t)st;
  bases[1] = (uint64_t)(uintptr_t)sh;
  bases[2] = (uint64_t)(uintptr_t)sl;
  bases[3] = (uint64_t)(uintptr_t)sb;
#pragma unroll
  for (int c = 0; c < 4; ++c) {
    bases[4 + c]  = (uint64_t)(uintptr_t)(rh + (size_t)c * HW_TOTAL);
    bases[8 + c]  = (uint64_t)(uintptr_t)(rt + (size_t)c * HW_TOTAL);
    bases[12 + c] = (uint64_t)(uintptr_t)(rb + (size_t)c * HW_TOTAL);
  }

  const uint32_t tile0 = blockIdx.x;  // tiles: tile0 + k*NBLOCKS, k = 0..3

  // Software pipeline, depth 2. Per-wave ASYNCcnt schedule (16 ops/tile):
  //   issue T0, T1                          -> cnt 32
  //   wait<=16 (T0 done), comp T0, issue T2 -> cnt<=32
  //   wait<=16 (T1 done), comp T1, issue T3 -> cnt<=32
  //   wait<=16 (T2 done), comp T2           -> cnt<=16
  //   wait<=0  (T3 done), comp T3
  issue_tile(bases, tile0 + 0u * NBLOCKS, 0u, tid, lds0);
  issue_tile(bases, tile0 + 1u * NBLOCKS, 1u, tid, lds0);

#pragma unroll
  for (uint32_t k = 0; k < TILES_PER_BLOCK; ++k) {
    if (k < TILES_PER_BLOCK - 1u)
      asm volatile("s_wait_asynccnt 0x10" ::: "memory");
    else
      asm volatile("s_wait_asynccnt 0x0" ::: "memory");

    compute_tile(smem, k & 1u, tile0 + k * NBLOCKS, tid, out);

    if (k + 2u < TILES_PER_BLOCK) {
      // WAR guard: our ds reads of this buffer must retire before the async
      // engine overwrites it (DScnt and ASYNCcnt retire independently).
      asm volatile("s_wait_dscnt 0x0" ::: "memory");
      issue_tile(bases, tile0 + (k + 2u) * NBLOCKS, k & 1u, tid, lds0);
    }
  }
}

extern "C" void kernel_launch(void* const* d_in, const int* in_sizes, int n_in,
                              void* d_out, int out_size, void* d_ws,
                              size_t ws_size, hipStream_t stream) {
  (void)in_sizes; (void)n_in; (void)out_size; (void)d_ws; (void)ws_size;
  points_gen_stream<<<dim3(NBLOCKS), dim3(THREADS), 0, stream>>>(
      (const float*)d_in[0], (const float*)d_in[1], (const float*)d_in[2],
      (const float*)d_in[3], (const float*)d_in[4], (const float*)d_in[5],
      (const float*)d_in[6], (float*)d_out);
}